// GCNAutoencoder_90512140796436
// MI455X (gfx1250) — compile-verified
//
#include <hip/hip_runtime.h>

// ---------------------------------------------------------------------------
// GCN autoencoder for MI455X (gfx1250): fp32 WMMA GEMMs + atomic COO SpMM.
// N = 100000 nodes (== 6250 * 16, so 16-row tiles cover exactly),
// E = 3.2M edges, dims 512 -> 32 -> 16 -> 32 -> 512.
// ---------------------------------------------------------------------------

typedef __attribute__((ext_vector_type(2))) float v2f;
typedef __attribute__((ext_vector_type(8))) float v8f;

// ---------------------------- zero-fill ------------------------------------
__global__ void zero_f32(float* __restrict__ p, long long n) {
  long long i = ((long long)blockIdx.x * blockDim.x + threadIdx.x) * 4;
  if (i + 3 < n) {
    *reinterpret_cast<float4*>(p + i) = make_float4(0.f, 0.f, 0.f, 0.f);
  } else {
    for (long long j = i; j < n; ++j) p[j] = 0.f;
  }
}

// ------------------------- dense GEMM via WMMA -----------------------------
// C[M, NCOL] = A[M, K] * B[K, NCOL], row-major, fp32.
// One wave32 computes one 16x16 tile of C with V_WMMA_F32_16X16X4_F32.
//
// A 16x4 tile layout (ISA 7.12.2): lane L holds row (L%16); its two A VGPRs
// hold K = k0 + 2*(L/16) and k0 + 2*(L/16)+1.
// B 4x16 tile: lane L holds col (L%16); its two B VGPRs hold rows
// K = k0 + 2*(L/16) and +1.
// C/D: VGPR v of lane L holds element (row = v + 8*(L/16), col = L%16).
template <int K, int NCOL>
__global__ void gemm16_wmma(const float* __restrict__ A,
                            const float* __restrict__ B,
                            float* __restrict__ C, int M) {
  const int gtid = blockIdx.x * blockDim.x + threadIdx.x;
  const int wave = gtid >> 5;
  const int lane = threadIdx.x & 31;
  constexpr int NT = NCOL / 16;          // column tiles
  const int tm = wave / NT;
  const int tn = wave % NT;
  if (tm * 16 >= M) return;              // wave-uniform guard (EXEC stays all-1)

  const int half = lane >> 4;            // 0: lanes 0-15, 1: lanes 16-31
  const int l16 = lane & 15;

  const float* __restrict__ arow = A + (size_t)(tm * 16 + l16) * K + 2 * half;
  const float* __restrict__ bcol = B + (size_t)(2 * half) * NCOL + (tn * 16 + l16);

  v8f acc = {};
#pragma unroll 4
  for (int k0 = 0; k0 < K; k0 += 4) {
    v2f a, b;
    a.x = arow[k0];
    a.y = arow[k0 + 1];
    b.x = bcol[(size_t)k0 * NCOL];
    b.y = bcol[(size_t)(k0 + 1) * NCOL];
    // 8 args: (neg_a, A, neg_b, B, c_mod, C, reuse_a, reuse_b)
    acc = __builtin_amdgcn_wmma_f32_16x16x4_f32(false, a, false, b,
                                                (short)0, acc, false, false);
  }

  float* __restrict__ cptr =
      C + (size_t)(tm * 16 + 8 * half) * NCOL + (size_t)tn * 16 + l16;
#pragma unroll
  for (int v = 0; v < 8; ++v) cptr[(size_t)v * NCOL] = acc[v];
}

// ------------------------------ COO SpMM -----------------------------------
// out[rows[e], :] += vals[e] * dense[cols[e], :]
// D/4 threads per edge, each moving one float4 (coalesced b128 gathers),
// scatter with global_atomic_add_f32.
template <int D>
__global__ void spmm_atomic(const int* __restrict__ rows,
                            const int* __restrict__ cols,
                            const float* __restrict__ vals,
                            const float* __restrict__ dense,
                            float* __restrict__ out, int nEdges) {
  constexpr int TPE = D / 4;             // threads per edge
  long long gtid = (long long)blockIdx.x * blockDim.x + threadIdx.x;
  int e = (int)(gtid / TPE);
  int c = (int)(gtid % TPE);
  if (e >= nEdges) return;

  const int r = rows[e];
  const int col = cols[e];
  const float v = vals[e];

  const float4 g =
      reinterpret_cast<const float4*>(dense + (size_t)col * D)[c];
  float* dst = out + (size_t)r * D + (size_t)c * 4;
  atomicAdd(dst + 0, v * g.x);
  atomicAdd(dst + 1, v * g.y);
  atomicAdd(dst + 2, v * g.z);
  atomicAdd(dst + 3, v * g.w);
}

// ---------------------------------------------------------------------------
extern "C" void kernel_launch(void* const* d_in, const int* in_sizes, int n_in,
                              void* d_out, int out_size, void* d_ws,
                              size_t ws_size, hipStream_t stream) {
  const float* x    = (const float*)d_in[0];
  const int*   rows = (const int*)d_in[1];
  const int*   cols = (const int*)d_in[2];
  const float* vals = (const float*)d_in[3];
  const float* W1   = (const float*)d_in[4];   // [512, 32]
  const float* W2   = (const float*)d_in[5];   // [32, 16]
  const float* W3   = (const float*)d_in[6];   // [16, 32]
  const float* W4   = (const float*)d_in[7];   // [32, 512]

  const int M = in_sizes[0] / 512;             // 100000 nodes
  const int E = in_sizes[1];                   // 3200000 edges

  float* out      = (float*)d_out;
  float* decoded2 = out;                       // [M, 512]  (output 0)
  float* encoded2 = out + (size_t)M * 512;     // [M, 16]   (output 1)

  // Workspace layout (buffers reused across stream-ordered phases):
  //   buf0 [M,32]  : support1, later support3
  //   buf1 [M,32]  : encoded1, later decoded1
  //   buf2 [M,16]  : support2
  //   buf3 [M,512] : support4
  float* ws   = (float*)d_ws;
  float* buf0 = ws;
  float* buf1 = buf0 + (size_t)M * 32;
  float* buf2 = buf1 + (size_t)M * 32;
  float* buf3 = buf2 + (size_t)M * 16;

  const int BLK = 256;
  const int mt = M / 16;                       // 6250 row tiles

  auto zblocks = [&](long long n) -> int {
    long long threads = (n + 3) / 4;
    return (int)((threads + BLK - 1) / BLK);
  };
  auto gblocks = [&](int nt) -> int {
    long long threads = (long long)mt * nt * 32;
    return (int)((threads + BLK - 1) / BLK);
  };
  auto sblocks = [&](int tpe) -> int {
    long long threads = (long long)E * tpe;
    return (int)((threads + BLK - 1) / BLK);
  };

  // Zero all atomic-scatter destinations (harness does not re-poison).
  zero_f32<<<zblocks((long long)M * 528), BLK, 0, stream>>>(
      out, (long long)M * 528);                               // decoded2+encoded2
  zero_f32<<<zblocks((long long)M * 32), BLK, 0, stream>>>(
      buf1, (long long)M * 32);                               // encoded1

  // Layer 1: support1 = x @ W1 ; encoded1 = spmm(support1)
  gemm16_wmma<512, 32><<<gblocks(2), BLK, 0, stream>>>(x, W1, buf0, M);
  spmm_atomic<32><<<sblocks(8), BLK, 0, stream>>>(rows, cols, vals, buf0, buf1, E);

  // Layer 2: support2 = encoded1 @ W2 ; encoded2 = spmm(support2)
  gemm16_wmma<32, 16><<<gblocks(1), BLK, 0, stream>>>(buf1, W2, buf2, M);
  spmm_atomic<16><<<sblocks(4), BLK, 0, stream>>>(rows, cols, vals, buf2, encoded2, E);

  // Re-zero buf1 for its decoded1 role.
  zero_f32<<<zblocks((long long)M * 32), BLK, 0, stream>>>(
      buf1, (long long)M * 32);

  // Layer 3: support3 = encoded2 @ W3 ; decoded1 = spmm(support3)
  gemm16_wmma<16, 32><<<gblocks(2), BLK, 0, stream>>>(encoded2, W3, buf0, M);
  spmm_atomic<32><<<sblocks(8), BLK, 0, stream>>>(rows, cols, vals, buf0, buf1, E);

  // Layer 4: support4 = decoded1 @ W4 ; decoded2 = spmm(support4)
  gemm16_wmma<32, 512><<<gblocks(32), BLK, 0, stream>>>(buf1, W4, buf3, M);
  spmm_atomic<512><<<sblocks(128), BLK, 0, stream>>>(rows, cols, vals, buf3, decoded2, E);
}